// GNN_7928509628453
// MI455X (gfx1250) — compile-verified
//
#include <hip/hip_runtime.h>
#include <hip/hip_bf16.h>

// ---------------------------------------------------------------------------
// GCN (2x GCNConv + mean-pool + FC) for MI455X (gfx1250, wave32).
// Memory-bound on the 800K-edge scatter (≈1.7GB traffic @ 23.3TB/s);
// dense GEMMs use V_WMMA_F32_16X16X4_F32 (only f32 WMMA shape on CDNA5).
// ---------------------------------------------------------------------------

typedef __attribute__((ext_vector_type(2))) float v2f;
typedef __attribute__((ext_vector_type(8))) float v8f;

#define FEAT 128
#define MTILE 32   // rows of A per block (2 row-tiles of 16)

// ---------------- zero fill ----------------
__global__ void gcn_zero_kernel(float* __restrict__ p, int n) {
    int i = blockIdx.x * blockDim.x + threadIdx.x;
    if (i < n) p[i] = 0.0f;
}

// ---------------- degree: deg[dst] += 1 per edge ----------------
__global__ void gcn_degree_kernel(const int* __restrict__ ei, float* __restrict__ deg, int E) {
    int e = blockIdx.x * blockDim.x + threadIdx.x;
    if (e < E) {
        int d = ei[E + e];          // row 1 of [2,E] = dst
        atomicAdd(deg + d, 1.0f);
    }
}

// ---------------- dinv[i] = (deg[i] + 1)^-1/2  (self loop adds 1) ----------------
__global__ void gcn_dinv_kernel(const float* __restrict__ deg, float* __restrict__ dinv, int n) {
    int i = blockIdx.x * blockDim.x + threadIdx.x;
    if (i < n) dinv[i] = rsqrtf(deg[i] + 1.0f);
}

// ---------------- GEMM: C[M x 128] = A[M x 128] @ W[128 x 128] via WMMA f32 ----------------
// 256 threads = 8 waves. Block covers MTILE=32 rows. Waves 0-3 -> rows 0-15,
// waves 4-7 -> rows 16-31; each wave computes two 16x16 output tiles.
__global__ __launch_bounds__(256) void gcn_gemm_wmma_kernel(const float* __restrict__ A,
                                                            const float* __restrict__ W,
                                                            float* __restrict__ C, int M) {
    __shared__ float Alds[MTILE * 132];   // stride 132 -> bank-conflict-free column reads

    const int tid = threadIdx.x;
    const int rowbase = blockIdx.x * MTILE;

    // Cooperative A-tile load: 32 rows x 32 float4 = 1024 float4, 4 per thread.
    for (int f4 = tid; f4 < MTILE * 32; f4 += 256) {
        int r = f4 >> 5;
        int c4 = f4 & 31;
        int row = rowbase + r;
        float4 v = make_float4(0.f, 0.f, 0.f, 0.f);
        if (row < M) v = *(const float4*)(A + (size_t)row * FEAT + c4 * 4);
        float* dst = &Alds[r * 132 + c4 * 4];
        dst[0] = v.x; dst[1] = v.y; dst[2] = v.z; dst[3] = v.w;
    }
    __syncthreads();

    const int wave = tid >> 5;
    const int lane = tid & 31;
    const int half = lane >> 4;     // 0: K={k,k+1}, 1: K={k+2,k+3}
    const int lid  = lane & 15;
    const int rt   = wave >> 2;     // row tile within block (0/1)
    const int wc   = wave & 3;      // which pair of column tiles
    const int arow = rt * 16 + lid;
    const int c0 = (wc * 2 + 0) * 16;
    const int c1 = (wc * 2 + 1) * 16;

    v8f acc0 = {};
    v8f acc1 = {};

    for (int k = 0; k < FEAT; k += 4) {
        const int ka = k + 2 * half;
        // A fragment (16x4): lane holds A[M=lid][ka], A[M=lid][ka+1]
        v2f a;
        a.x = Alds[arow * 132 + ka];
        a.y = Alds[arow * 132 + ka + 1];
        // B fragments (4x16): lane holds W[ka][col], W[ka+1][col]
        v2f b0, b1;
        b0.x = W[ka * FEAT + c0 + lid];
        b0.y = W[(ka + 1) * FEAT + c0 + lid];
        b1.x = W[ka * FEAT + c1 + lid];
        b1.y = W[(ka + 1) * FEAT + c1 + lid];
        acc0 = __builtin_amdgcn_wmma_f32_16x16x4_f32(false, a, false, b0, (short)0, acc0, false, false);
        acc1 = __builtin_amdgcn_wmma_f32_16x16x4_f32(false, a, false, b1, (short)0, acc1, false, false);
    }

    // D layout: VGPR r -> M = r + 8*half, N = lid
    #pragma unroll
    for (int r = 0; r < 8; ++r) {
        int row = rowbase + rt * 16 + r + 8 * half;
        if (row < M) {
            C[(size_t)row * FEAT + c0 + lid] = acc0[r];
            C[(size_t)row * FEAT + c1 + lid] = acc1[r];
        }
    }
}

// ---------------- edge scatter: aggr[dst] += t[src] * dinv[src]*dinv[dst] ----------------
// One wave per edge; each lane moves a float4 (32*4 = 128 feats).
__global__ void gcn_edge_scatter_kernel(const float* __restrict__ t,
                                        const float* __restrict__ dinv,
                                        const int* __restrict__ ei,
                                        float* __restrict__ aggr, int E) {
    int wid = (blockIdx.x * blockDim.x + threadIdx.x) >> 5;
    int lane = threadIdx.x & 31;
    if (wid >= E) return;
    int s = ei[wid];
    int d = ei[E + wid];
    float nrm = dinv[s] * dinv[d];
    float4 v = *(const float4*)(t + (size_t)s * FEAT + lane * 4);
    float* dp = aggr + (size_t)d * FEAT + lane * 4;
    atomicAdd(dp + 0, v.x * nrm);
    atomicAdd(dp + 1, v.y * nrm);
    atomicAdd(dp + 2, v.z * nrm);
    atomicAdd(dp + 3, v.w * nrm);
}

// ---------------- finalize: aggr = relu(aggr + dinv^2 * t + bias) ----------------
__global__ void gcn_finalize_kernel(float* __restrict__ aggr, const float* __restrict__ t,
                                    const float* __restrict__ dinv, const float* __restrict__ bias,
                                    int n) {
    int i = blockIdx.x * blockDim.x + threadIdx.x;
    if (i >= n * FEAT) return;
    int node = i >> 7;
    int f = i & (FEAT - 1);
    float di = dinv[node];
    float v = aggr[i] + di * di * t[i] + bias[f];
    aggr[i] = v > 0.0f ? v : 0.0f;
}

// ---------------- mean-pool accumulation ----------------
__global__ void gcn_pool_kernel(const float* __restrict__ h, const int* __restrict__ batch,
                                float* __restrict__ sums, float* __restrict__ cnt, int n) {
    int i = blockIdx.x * blockDim.x + threadIdx.x;
    if (i >= n * FEAT) return;
    int node = i >> 7;
    int f = i & (FEAT - 1);
    int g = batch[node];
    atomicAdd(&sums[g * FEAT + f], h[i]);
    if (f == 0) atomicAdd(&cnt[g], 1.0f);
}

// ---------------- FC: out[g,l] = (sums[g]/max(cnt,1)) . Wfc[:,l] + bfc[l] ----------------
__global__ void gcn_fc_kernel(const float* __restrict__ sums, const float* __restrict__ cnt,
                              const float* __restrict__ Wfc, const float* __restrict__ bfc,
                              float* __restrict__ out, int G, int L) {
    int i = blockIdx.x * blockDim.x + threadIdx.x;
    if (i >= G * L) return;
    int g = i / L;
    int l = i - g * L;
    float c = cnt[g];
    c = c > 1.0f ? c : 1.0f;
    float s = 0.0f;
    for (int k = 0; k < FEAT; ++k) s += sums[g * FEAT + k] * Wfc[k * L + l];
    out[i] = s / c + bfc[l];
}

extern "C" void kernel_launch(void* const* d_in, const int* in_sizes, int n_in,
                              void* d_out, int out_size, void* d_ws, size_t ws_size,
                              hipStream_t stream) {
    const float* x    = (const float*)d_in[0];
    const int*   ei   = (const int*)d_in[1];
    const int*   batch= (const int*)d_in[2];
    const float* W1   = (const float*)d_in[3];
    const float* b1   = (const float*)d_in[4];
    const float* W2   = (const float*)d_in[5];
    const float* b2   = (const float*)d_in[6];
    const float* Wfc  = (const float*)d_in[7];
    const float* bfc  = (const float*)d_in[8];
    float* out = (float*)d_out;

    const int N = in_sizes[0] / FEAT;      // 50000 nodes
    const int E = in_sizes[1] / 2;         // 800000 edges
    const int G = 128;                     // graphs
    const int L = in_sizes[8];             // label dim (7)

    // workspace carve (floats)
    float* ws   = (float*)d_ws;
    float* tbuf = ws;                       // N*FEAT  (GEMM output / messages)
    float* abuf = tbuf + (size_t)N * FEAT;  // N*FEAT  (aggregation / hidden)
    float* deg  = abuf + (size_t)N * FEAT;  // N
    float* dinv = deg + N;                  // N
    float* sums = dinv + N;                 // G*FEAT
    float* cnt  = sums + (size_t)G * FEAT;  // G

    const int nf = N * FEAT;
    dim3 b256(256);
    int gridNF   = (nf + 255) / 256;
    int gridE    = (E + 255) / 256;
    int gridEw   = (E * 32 + 255) / 256;   // one wave per edge
    int gridN    = (N + 255) / 256;
    int gridGemm = (N + MTILE - 1) / MTILE;

    // degree + normalization
    gcn_zero_kernel<<<gridN, b256, 0, stream>>>(deg, N);
    gcn_degree_kernel<<<gridE, b256, 0, stream>>>(ei, deg, E);
    gcn_dinv_kernel<<<gridN, b256, 0, stream>>>(deg, dinv, N);

    // ---- layer 1 ----
    gcn_gemm_wmma_kernel<<<gridGemm, b256, 0, stream>>>(x, W1, tbuf, N);
    gcn_zero_kernel<<<gridNF, b256, 0, stream>>>(abuf, nf);
    gcn_edge_scatter_kernel<<<gridEw, b256, 0, stream>>>(tbuf, dinv, ei, abuf, E);
    gcn_finalize_kernel<<<gridNF, b256, 0, stream>>>(abuf, tbuf, dinv, b1, N);   // h1 in abuf

    // ---- layer 2 ----
    gcn_gemm_wmma_kernel<<<gridGemm, b256, 0, stream>>>(abuf, W2, tbuf, N);      // t2 in tbuf
    gcn_zero_kernel<<<gridNF, b256, 0, stream>>>(abuf, nf);                      // h1 dead now
    gcn_edge_scatter_kernel<<<gridEw, b256, 0, stream>>>(tbuf, dinv, ei, abuf, E);
    gcn_finalize_kernel<<<gridNF, b256, 0, stream>>>(abuf, tbuf, dinv, b2, N);   // h2 in abuf

    // ---- pool + fc ----
    gcn_zero_kernel<<<(G * FEAT + G + 255) / 256, b256, 0, stream>>>(sums, G * FEAT + G); // sums+cnt contiguous
    gcn_pool_kernel<<<gridNF, b256, 0, stream>>>(abuf, batch, sums, cnt, N);
    gcn_fc_kernel<<<(G * L + 255) / 256, b256, 0, stream>>>(sums, cnt, Wfc, bfc, out, G, L);
}